// SurfaceNet2_76622216561215
// MI455X (gfx1250) — compile-verified
//
#include <hip/hip_runtime.h>
#include <hip/hip_bf16.h>

typedef __attribute__((ext_vector_type(2))) float v2f;
typedef __attribute__((ext_vector_type(8))) float v8f;
typedef unsigned int u32x4 __attribute__((ext_vector_type(4)));
typedef int i32x8 __attribute__((ext_vector_type(8)));
typedef int i32x4 __attribute__((ext_vector_type(4)));

#define NB   2
#define NPT  16384      // N (query points)
#define NS   4096       // S (source points)
#define D1C  128
#define D2C  256
#define ICC  384        // D1 + D2
#define OCC  128
#define PTOT (NB * NPT) // 32768 positions for BN stats
#define EPS_BN 1e-5f

#define KT    32        // K-panel rows staged per stage
#define LDST  72        // padded LDS row stride in floats (64 data + 8 pad dwords)

#if defined(__has_builtin)
#if __has_builtin(__builtin_amdgcn_tensor_load_to_lds) && \
    __has_builtin(__builtin_amdgcn_s_wait_tensorcnt)
#define USE_TDM 1
#endif
#endif
#ifndef USE_TDM
#define USE_TDM 0
#endif

// ---------------------------------------------------------------------------
// WMMA wrapper: D = A(16x4 f32) * B(4x16 f32) + C(16x16 f32)
// ---------------------------------------------------------------------------
__device__ __forceinline__ v8f wmma4(v2f a, v2f b, v8f c) {
  return __builtin_amdgcn_wmma_f32_16x16x4_f32(
      /*neg_a=*/false, a, /*neg_b=*/false, b,
      /*c_mod=*/(short)0, c, /*reuse_a=*/false, /*reuse_b=*/false);
}

// ---------------------------------------------------------------------------
// Stage a KT x 64 f32 tile (row stride NPT) into LDS with row stride LDST.
// TDM path: one wave issues tensor_load_to_lds with a D# built per the CDNA5
// ISA (CDNA5 ISA):
//   group0: count=1 | lds_addr | global_addr(tile start) | type=2
//   group1: data_size=4B, pad_enable, pad_interval=64dw(code5),
//           pad_amount=8dw(code7), tensor_dim0=NPT, tensor_dim1=KT,
//           tile_dim0=64, tile_dim1=KT, tensor_dim0_stride=NPT
// Caller must __syncthreads() after this returns.
// ---------------------------------------------------------------------------
__device__ __forceinline__ void stage_tile(const float* __restrict__ base, // chunk row0, col0
                                           float* __restrict__ sX,
                                           int n0, int tid, int wave) {
#if USE_TDM
  if (wave == 0) {
    unsigned long long ga = (unsigned long long)(uintptr_t)(base + n0);
    unsigned lds_off = (unsigned)(uintptr_t)sX;   // LDS aperture: low 32 bits = offset
    u32x4 g0 = { 1u,                               // count=1, user descriptor
                 lds_off,
                 (unsigned)(ga & 0xFFFFFFFFull),
                 (unsigned)((ga >> 32) & 0x01FFFFFFull) | 0x80000000u };  // type=2
    i32x8 g1 = { (int)((2u << 16) | (1u << 20) | (5u << 22) | (7u << 25)),
                 (int)(((unsigned)NPT & 0xFFFFu) << 16),   // tensor_dim0[15:0]
                 (int)(((unsigned)KT  & 0xFFFFu) << 16),   // tensor_dim1[15:0]
                 (int)(64u << 16),                          // tile_dim0 = 64
                 (int)(KT),                                 // tile_dim1 = KT, tile_dim2 = 0
                 (int)(NPT),                                // tensor_dim0_stride[31:0]
                 0, 0 };
    i32x4 g2 = { 0, 0, 0, 0 };
    i32x4 g3 = { 0, 0, 0, 0 };
#if __clang_major__ >= 23
    i32x8 g4 = { 0, 0, 0, 0, 0, 0, 0, 0 };
    __builtin_amdgcn_tensor_load_to_lds(g0, g1, g2, g3, g4, 0);
#else
    __builtin_amdgcn_tensor_load_to_lds(g0, g1, g2, g3, 0);
#endif
    __builtin_amdgcn_s_wait_tensorcnt(0);
  }
#else
  // Fallback: cooperative float4 staging by all 256 threads.
  for (int t = tid; t < KT * 16; t += 256) {
    const int row = t >> 4;
    const int c4  = (t & 15) << 2;
    *(float4*)(sX + row * LDST + c4) =
        *(const float4*)(base + (size_t)row * NPT + n0 + c4);
  }
#endif
}

// ---------------------------------------------------------------------------
// Kernel 0: zero the per-channel BN partial-sum accumulators (3 convs x 2x128)
// ---------------------------------------------------------------------------
__global__ void k_zero_stats(float* __restrict__ st) {
  st[threadIdx.x] = 0.0f;   // launched with 768 threads
}

// ---------------------------------------------------------------------------
// Kernel 1: brute-force 3-NN + inverse-distance interpolation.
// xyz2 (with |p|^2) staged in 64KB LDS; each thread owns one query point.
// Writes interp channel-major: itp[b][c][n]  (c in 0..255)
// ---------------------------------------------------------------------------
__global__ void __launch_bounds__(256)
k_knn_interp(const float* __restrict__ xyz1, const float* __restrict__ xyz2,
             const float* __restrict__ p2, float* __restrict__ itp) {
  __shared__ float4 sQ[NS];  // 64 KB

  const int b   = blockIdx.y;
  const int tid = threadIdx.x;

  const float* x2b = xyz2 + (size_t)b * NS * 3;
  for (int i = tid; i < NS; i += 256) {
    float x = x2b[i * 3 + 0];
    float y = x2b[i * 3 + 1];
    float z = x2b[i * 3 + 2];
    sQ[i] = make_float4(x, y, z, x * x + y * y + z * z);
  }
  __syncthreads();

  const int n = blockIdx.x * 256 + tid;
  const float* x1 = xyz1 + ((size_t)b * NPT + n) * 3;
  const float px = x1[0], py = x1[1], pz = x1[2];
  const float n1 = px * px + py * py + pz * pz;

  float d0 = 3.4e38f, d1 = 3.4e38f, d2 = 3.4e38f;
  int   i0 = 0, i1 = 0, i2 = 0;

#pragma unroll 4
  for (int s = 0; s < NS; ++s) {
    float4 q = sQ[s];
    float d = n1 + q.w - 2.0f * (px * q.x + py * q.y + pz * q.z);
    if (d < d2) {
      if (d < d1) {
        d2 = d1; i2 = i1;
        if (d < d0) { d1 = d0; i1 = i0; d0 = d; i0 = s; }
        else        { d1 = d;  i1 = s; }
      } else { d2 = d; i2 = s; }
    }
  }

  float w0 = 1.0f / (d0 + 1e-8f);
  float w1 = 1.0f / (d1 + 1e-8f);
  float w2 = 1.0f / (d2 + 1e-8f);
  const float wn = 1.0f / (w0 + w1 + w2);
  w0 *= wn; w1 *= wn; w2 *= wn;

  const float* p2b = p2 + (size_t)b * D2C * NS;
  float* ob = itp + (size_t)b * D2C * NPT + n;
#pragma unroll 4
  for (int c = 0; c < D2C; ++c) {
    const float* row = p2b + (size_t)c * NS;
    ob[(size_t)c * NPT] = w0 * row[i0] + w1 * row[i1] + w2 * row[i2];
  }
}

// ---------------------------------------------------------------------------
// Shared GEMM epilogue: +bias, store pre-BN z, per-channel sum/sumsq atomics.
// C-layout: VGPR r of acc[j] holds channel m0+r (lanes 0-15) / m0+r+8
// (lanes 16-31), column n0 + 16*j + (lane&15).
// ---------------------------------------------------------------------------
__device__ __forceinline__ void gemm_epilogue(v8f* acc, const float* __restrict__ bias,
                                              float* __restrict__ zb,
                                              float* __restrict__ sum, float* __restrict__ sq,
                                              int m0, int n0, int lane) {
  const int hl   = lane & 15;
  const int mAdd = (lane < 16) ? 0 : 8;
#pragma unroll
  for (int r = 0; r < 8; ++r) {
    const int m  = m0 + r + mAdd;
    const float bv = bias[m];
    float ps = 0.0f, pq = 0.0f;
#pragma unroll
    for (int j = 0; j < 4; ++j) {
      float z = acc[j][r] + bv;
      zb[(size_t)m * NPT + (n0 + j * 16 + hl)] = z;
      ps += z;
      pq += z * z;
    }
    // reduce across the 16-lane half that shares channel m
#pragma unroll
    for (int off = 8; off > 0; off >>= 1) {
      ps += __shfl_xor(ps, off, 16);
      pq += __shfl_xor(pq, off, 16);
    }
    if (hl == 0) {
      atomicAdd(&sum[m], ps);
      atomicAdd(&sq[m], pq);
    }
  }
}

// ---------------------------------------------------------------------------
// Kernel 2: fuse conv  z1[b,o,n] = sum_c W[o,c]*concat(points1,interp)[b,c,n]+b[o]
// K=384. TDM stages 32x64 input panels into LDS; 8 waves x (16M x 64N) WMMA.
// ---------------------------------------------------------------------------
__global__ void __launch_bounds__(256)
k_gemm_fuse(const float* __restrict__ p1, const float* __restrict__ itp,
            const float* __restrict__ W, const float* __restrict__ bias,
            float* __restrict__ zout, float* __restrict__ sum, float* __restrict__ sq) {
  __shared__ float sX[KT * LDST];  // 9216 B

  const int b    = blockIdx.y;
  const int n0   = blockIdx.x * 64;
  const int tid  = threadIdx.x;
  const int lane = tid & 31;
  const int wave = tid >> 5;
  const int m0   = wave * 16;
  const int hl   = lane & 15;
  const int koff = (lane < 16) ? 0 : 2;

  const float* p1b = p1 + (size_t)b * D1C * NPT;
  const float* itb = itp + (size_t)b * D2C * NPT;

  v8f acc[4] = {v8f(0.0f), v8f(0.0f), v8f(0.0f), v8f(0.0f)};

  for (int k0 = 0; k0 < ICC; k0 += KT) {
    __syncthreads();   // previous panel fully consumed
    // KT=32 divides the concat boundary (128): chunk is wholly in one source.
    const float* base = (k0 < D1C) ? (p1b + (size_t)k0 * NPT)
                                   : (itb + (size_t)(k0 - D1C) * NPT);
    stage_tile(base, sX, n0, tid, wave);
    __syncthreads();   // panel visible to all waves

#pragma unroll
    for (int kk = 0; kk < KT; kk += 4) {
      const int k = kk + koff;
      const float* wr = W + (size_t)(m0 + hl) * ICC + k0 + k;
      v2f a;
      a.x = wr[0];
      a.y = wr[1];
      const float* l0 = sX + k * LDST;
      const float* l1 = l0 + LDST;
#pragma unroll
      for (int j = 0; j < 4; ++j) {
        const int col = j * 16 + hl;
        v2f bf;
        bf.x = l0[col];
        bf.y = l1[col];
        acc[j] = wmma4(a, bf, acc[j]);
      }
    }
  }

  gemm_epilogue(acc, bias, zout + (size_t)b * OCC * NPT, sum, sq, m0, n0, lane);
}

// ---------------------------------------------------------------------------
// Kernel 3: residual conv  z_out = W * relu((z_in - mean)*scale) + b, K = 128.
// Raw z staged via TDM; BN+ReLU applied on LDS-read values.
// ---------------------------------------------------------------------------
__global__ void __launch_bounds__(256)
k_gemm_res(const float* __restrict__ zin, const float* __restrict__ W,
           const float* __restrict__ bias, const float* __restrict__ mean,
           const float* __restrict__ scale, float* __restrict__ zout,
           float* __restrict__ sum, float* __restrict__ sq) {
  __shared__ float sX[KT * LDST];
  __shared__ float sMu[OCC];
  __shared__ float sSc[OCC];

  const int b    = blockIdx.y;
  const int n0   = blockIdx.x * 64;
  const int tid  = threadIdx.x;
  const int lane = tid & 31;
  const int wave = tid >> 5;
  const int m0   = wave * 16;
  const int hl   = lane & 15;
  const int koff = (lane < 16) ? 0 : 2;

  if (tid < OCC) {
    sMu[tid] = mean[tid];
    sSc[tid] = scale[tid];
  }

  const float* zi = zin + (size_t)b * OCC * NPT;

  v8f acc[4] = {v8f(0.0f), v8f(0.0f), v8f(0.0f), v8f(0.0f)};

  for (int k0 = 0; k0 < OCC; k0 += KT) {
    __syncthreads();   // prev panel consumed (also covers sMu/sSc init)
    stage_tile(zi + (size_t)k0 * NPT, sX, n0, tid, wave);
    __syncthreads();

#pragma unroll
    for (int kk = 0; kk < KT; kk += 4) {
      const int k  = kk + koff;
      const int kg = k0 + k;
      const float* wr = W + (size_t)(m0 + hl) * OCC + kg;
      v2f a;
      a.x = wr[0];
      a.y = wr[1];
      const float mu0 = sMu[kg],     sc0 = sSc[kg];
      const float mu1 = sMu[kg + 1], sc1 = sSc[kg + 1];
      const float* l0 = sX + k * LDST;
      const float* l1 = l0 + LDST;
#pragma unroll
      for (int j = 0; j < 4; ++j) {
        const int col = j * 16 + hl;
        v2f bf;
        bf.x = fmaxf((l0[col] - mu0) * sc0, 0.0f);
        bf.y = fmaxf((l1[col] - mu1) * sc1, 0.0f);
        acc[j] = wmma4(a, bf, acc[j]);
      }
    }
  }

  gemm_epilogue(acc, bias, zout + (size_t)b * OCC * NPT, sum, sq, m0, n0, lane);
}

// ---------------------------------------------------------------------------
// Kernel 4: finalize BN stats -> mean, rsqrt(var+eps)   (biased var, /P)
// ---------------------------------------------------------------------------
__global__ void k_bn_stats(const float* __restrict__ sum, const float* __restrict__ sq,
                           float* __restrict__ mean, float* __restrict__ scale, float invP) {
  const int c = threadIdx.x;  // 128 threads
  const float m = sum[c] * invP;
  const float v = sq[c] * invP - m * m;
  mean[c]  = m;
  scale[c] = rsqrtf(v + EPS_BN);
}

// ---------------------------------------------------------------------------
// Kernel 5: out = relu( bn3(z3) + relu(bn1(z1)) )   (vectorized x4)
// ---------------------------------------------------------------------------
__global__ void __launch_bounds__(256)
k_final(const float* __restrict__ z1, const float* __restrict__ z3,
        const float* __restrict__ m1, const float* __restrict__ s1,
        const float* __restrict__ m3, const float* __restrict__ s3,
        float* __restrict__ out) {
  const size_t i = ((size_t)blockIdx.x * 256 + threadIdx.x) * 4;
  const int o = (int)((i >> 14) & 127);  // NPT = 2^14
  const float mu1 = m1[o], sc1 = s1[o];
  const float mu3 = m3[o], sc3 = s3[o];
  const float4 a = *(const float4*)(z1 + i);
  const float4 h = *(const float4*)(z3 + i);
  float4 r;
  r.x = fmaxf((h.x - mu3) * sc3 + fmaxf((a.x - mu1) * sc1, 0.0f), 0.0f);
  r.y = fmaxf((h.y - mu3) * sc3 + fmaxf((a.y - mu1) * sc1, 0.0f), 0.0f);
  r.z = fmaxf((h.z - mu3) * sc3 + fmaxf((a.z - mu1) * sc1, 0.0f), 0.0f);
  r.w = fmaxf((h.w - mu3) * sc3 + fmaxf((a.w - mu1) * sc1, 0.0f), 0.0f);
  *(float4*)(out + i) = r;
}

// ---------------------------------------------------------------------------
extern "C" void kernel_launch(void* const* d_in, const int* in_sizes, int n_in,
                              void* d_out, int out_size, void* d_ws, size_t ws_size,
                              hipStream_t stream) {
  const float* xyz1    = (const float*)d_in[0];
  const float* xyz2    = (const float*)d_in[1];
  const float* points1 = (const float*)d_in[2];
  const float* points2 = (const float*)d_in[3];
  const float* fuse_w  = (const float*)d_in[4];
  const float* fuse_b  = (const float*)d_in[5];
  const float* res_w1  = (const float*)d_in[6];
  const float* res_b1  = (const float*)d_in[7];
  const float* res_w2  = (const float*)d_in[8];
  const float* res_b2  = (const float*)d_in[9];
  float* out = (float*)d_out;

  float* ws  = (float*)d_ws;
  float* itp = ws;                                  // B*256*N = 8388608 floats
  float* z1  = itp + (size_t)NB * D2C * NPT;        // B*128*N
  float* z2  = z1 + (size_t)NB * OCC * NPT;
  float* z3  = z2 + (size_t)NB * OCC * NPT;
  float* st  = z3 + (size_t)NB * OCC * NPT;
  float* sum1 = st + 0,    *sq1 = st + 128;
  float* sum2 = st + 256,  *sq2 = st + 384;
  float* sum3 = st + 512,  *sq3 = st + 640;
  float* mean1 = st + 768,  *scl1 = st + 896;
  float* mean2 = st + 1024, *scl2 = st + 1152;
  float* mean3 = st + 1280, *scl3 = st + 1408;

  const float invP = 1.0f / (float)PTOT;

  k_zero_stats<<<1, 768, 0, stream>>>(st);
  k_knn_interp<<<dim3(NPT / 256, NB), 256, 0, stream>>>(xyz1, xyz2, points2, itp);
  k_gemm_fuse<<<dim3(NPT / 64, NB), 256, 0, stream>>>(points1, itp, fuse_w, fuse_b,
                                                      z1, sum1, sq1);
  k_bn_stats<<<1, 128, 0, stream>>>(sum1, sq1, mean1, scl1, invP);
  k_gemm_res<<<dim3(NPT / 64, NB), 256, 0, stream>>>(z1, res_w1, res_b1, mean1, scl1,
                                                     z2, sum2, sq2);
  k_bn_stats<<<1, 128, 0, stream>>>(sum2, sq2, mean2, scl2, invP);
  k_gemm_res<<<dim3(NPT / 64, NB), 256, 0, stream>>>(z2, res_w2, res_b2, mean2, scl2,
                                                     z3, sum3, sq3);
  k_bn_stats<<<1, 128, 0, stream>>>(sum3, sq3, mean3, scl3, invP);
  k_final<<<(NB * OCC * NPT) / (4 * 256), 256, 0, stream>>>(z1, z3, mean1, scl1,
                                                            mean3, scl3, out);
}